// DurationAlignmentModule_69269232550091
// MI455X (gfx1250) — compile-verified
//
#include <hip/hip_runtime.h>
#include <hip/hip_bf16.h>

typedef float v2f __attribute__((ext_vector_type(2)));
typedef float v8f __attribute__((ext_vector_type(8)));

#define Bn 32
#define Tn 1024
#define Dn 512
#define Hn 256
#define BN_EPS 1e-5f

// ---------------------------------------------------------------------------
// Weight repack:  w[h][d][kk]  ->  wt[kk][d>>2][h][d&3]
// Makes each WMMA B-fragment a single coalesced 8-byte load:
//   lane needs Wt[k, n], Wt[k+1, n] with (k&3) == kOff in {0,2}  ==>  v2f at
//   wt[((kk*(CIN/4) + k/4)*256 + n)*4 + kOff]
// ---------------------------------------------------------------------------
template <int CIN>
__global__ __launch_bounds__(256) void repack_w_kernel(
    const float* __restrict__ w, float* __restrict__ wt) {
  const int i = blockIdx.x * 256 + threadIdx.x;  // flat over Hn*CIN*3
  const int kk = i % 3;
  const int rest = i / 3;
  const int d = rest % CIN;
  const int h = rest / CIN;
  wt[(((size_t)kk * (CIN >> 2) + (d >> 2)) * Hn + h) * 4 + (d & 3)] = w[i];
}

// ---------------------------------------------------------------------------
// conv1d(K=3, pad=1) as GEMM + BatchNorm + ReLU via V_WMMA_F32_16X16X4_F32.
//   y[b,t,h] = relu( BN( sum_{kk,d} x[b, t+kk-1, d] * w[h,d,kk] + bias[h] ) )
// x: [B, T, CIN] (row = GEMM A row), y: [B, T, 256].
// One block = 32 rows (2 M-tiles) x 256 cols; 8 waves; each wave owns a
// 2x2 tile block (2 M-tiles x 2 N-tiles) so every A/B fragment feeds 2 WMMAs.
// ---------------------------------------------------------------------------
template <int CIN>
__global__ __launch_bounds__(256) void conv_bn_relu_wmma(
    const float* __restrict__ x, const float* __restrict__ wt,
    const float* __restrict__ bias, const float* __restrict__ gamma,
    const float* __restrict__ beta, const float* __restrict__ mean,
    const float* __restrict__ var, float* __restrict__ y) {
  constexpr int KCH = 128;                 // K-chunk staged in LDS
  constexpr int LSTR = KCH + 4;            // padded row -> conflict-free banks
  __shared__ float a_lds[32 * LSTR];

  const int row0 = blockIdx.x * 32;        // 32 rows of (b,t); T%32==0
  const int b = row0 / Tn;
  const int t0 = row0 % Tn;
  const int tid = threadIdx.x;
  const int wave = tid >> 5;
  const int lane = tid & 31;
  const int n0 = wave * 32;                // wave's two N-tiles: n0, n0+16
  const int mrow = lane & 15;              // A-frag M index
  const int kOff = (lane >> 4) << 1;       // half-wave K offset (0 or 2)

  v8f acc00 = {0.f, 0.f, 0.f, 0.f, 0.f, 0.f, 0.f, 0.f};
  v8f acc01 = acc00, acc10 = acc00, acc11 = acc00;

  for (int kk = 0; kk < 3; ++kk) {
    for (int kc = 0; kc < CIN; kc += KCH) {
      // stage shifted A patch, float4 loads / b128 LDS stores (zero at edges)
      for (int i = tid; i < 32 * (KCH / 4); i += 256) {
        const int r = i >> 5;              // KCH/4 == 32 float4 per row
        const int c4 = i & 31;
        const int tt = t0 + r + kk - 1;
        float4 v = {0.f, 0.f, 0.f, 0.f};
        if (tt >= 0 && tt < Tn)
          v = *(const float4*)&x[((size_t)b * Tn + tt) * CIN + kc + c4 * 4];
        *(float4*)&a_lds[r * LSTR + c4 * 4] = v;
      }
      __syncthreads();

      const float* wq0 = wt + (((size_t)kk * (CIN >> 2) + (kc >> 2)) * Hn +
                               (n0 + mrow)) * 4 + kOff;
      const float* wq1 = wq0 + 64;         // +16 columns * 4
#pragma unroll
      for (int ks = 0; ks < KCH; ks += 4) {
        const int kb = ks + kOff;
        v2f a0 = *(const v2f*)&a_lds[mrow * LSTR + kb];
        v2f a1 = *(const v2f*)&a_lds[(16 + mrow) * LSTR + kb];
        v2f b0 = *(const v2f*)&wq0[(ks >> 2) * (Hn * 4)];
        v2f b1 = *(const v2f*)&wq1[(ks >> 2) * (Hn * 4)];
        acc00 = __builtin_amdgcn_wmma_f32_16x16x4_f32(false, a0, false, b0,
                                                      (short)0, acc00, false, false);
        acc01 = __builtin_amdgcn_wmma_f32_16x16x4_f32(false, a0, false, b1,
                                                      (short)0, acc01, false, false);
        acc10 = __builtin_amdgcn_wmma_f32_16x16x4_f32(false, a1, false, b0,
                                                      (short)0, acc10, false, false);
        acc11 = __builtin_amdgcn_wmma_f32_16x16x4_f32(false, a1, false, b1,
                                                      (short)0, acc11, false, false);
      }
      __syncthreads();
    }
  }

  // Epilogue: fold conv bias into BN: z = conv*s + c,
  //   s = g*rsqrt(v+eps), c = (bias-mean)*s + beta ; then ReLU.
  // C layout: lanes0-15 -> M=r, lanes16-31 -> M=r+8; N = lane&15.
  const int h0 = n0 + mrow;
  const int h1 = h0 + 16;
  const float s0 = gamma[h0] * rsqrtf(var[h0] + BN_EPS);
  const float c0 = (bias[h0] - mean[h0]) * s0 + beta[h0];
  const float s1 = gamma[h1] * rsqrtf(var[h1] + BN_EPS);
  const float c1 = (bias[h1] - mean[h1]) * s1 + beta[h1];
  const int mb0 = row0 + ((lane >> 4) << 3);
  const int mb1 = mb0 + 16;
#pragma unroll
  for (int r = 0; r < 8; ++r) {
    const size_t off0 = (size_t)(mb0 + r) * Hn;
    const size_t off1 = (size_t)(mb1 + r) * Hn;
    y[off0 + h0] = fmaxf(fmaf(acc00[r], s0, c0), 0.f);
    y[off0 + h1] = fmaxf(fmaf(acc01[r], s1, c1), 0.f);
    y[off1 + h0] = fmaxf(fmaf(acc10[r], s0, c0), 0.f);
    y[off1 + h1] = fmaxf(fmaf(acc11[r], s1, c1), 0.f);
  }
}

// ---------------------------------------------------------------------------
// proj (1x1 conv 256->1) + loss terms. One wave per row at a time with
// coalesced float4 loads + shuffle reduction; fixed-order partial sums.
// ---------------------------------------------------------------------------
__global__ __launch_bounds__(256) void proj_loss_kernel(
    const float* __restrict__ act2, const float* __restrict__ pw,
    const float* __restrict__ pb, const int* __restrict__ dur,
    float* __restrict__ out_dur, float* __restrict__ partials) {
  __shared__ float red[8];
  const int wave = threadIdx.x >> 5;
  const int lane = threadIdx.x & 31;
  const int gwave = blockIdx.x * 8 + wave;   // 128 blocks * 8 waves = 1024
  const float4* p4 = (const float4*)pw;      // 64 float4
  const float4 q0 = p4[lane];
  const float4 q1 = p4[lane + 32];
  float wsum = 0.f;
  for (int r = 0; r < 32; ++r) {             // 32 rows per wave
    const int row = gwave * 32 + r;
    const float4* a = (const float4*)(act2 + (size_t)row * Hn);
    const float4 x0 = a[lane];
    const float4 x1 = a[lane + 32];
    float s = x0.x * q0.x + x0.y * q0.y + x0.z * q0.z + x0.w * q0.w +
              x1.x * q1.x + x1.y * q1.y + x1.z * q1.z + x1.w * q1.w;
    for (int o = 16; o > 0; o >>= 1) s += __shfl_down(s, o, 32);
    if (lane == 0) {
      const float ld = s + pb[0];
      const float df = (float)dur[row];
      out_dur[row] = df;
      const float diff = logf(expf(ld) + 1.f) - logf(df + 1.f);
      wsum += diff * diff;
    }
  }
  if (lane == 0) red[wave] = wsum;
  __syncthreads();
  if (threadIdx.x == 0) {
    float t = 0.f;
    for (int i = 0; i < 8; ++i) t += red[i];
    partials[blockIdx.x] = t;
  }
}

__global__ void loss_final_kernel(const float* __restrict__ partials, int n,
                                  float* __restrict__ out) {
  if (threadIdx.x == 0) {
    float s = 0.f;
    for (int i = 0; i < n; ++i) s += partials[i];  // fixed order: deterministic
    out[0] = s / (float)(Bn * Tn);
  }
}

// ---------------------------------------------------------------------------
// Length regulator: zero expanded region, per-batch cumsum, disjoint scatter.
// ---------------------------------------------------------------------------
__global__ void zero_kernel(float* __restrict__ p, size_t n) {
  size_t i = (size_t)blockIdx.x * blockDim.x + threadIdx.x;
  const size_t stride = (size_t)gridDim.x * blockDim.x;
  for (; i < n; i += stride) p[i] = 0.f;
}

__global__ void cumsum_kernel(const int* __restrict__ dur,
                              int* __restrict__ cum) {
  const int b = threadIdx.x;
  if (b < Bn) {
    int c = 0;
    for (int t = 0; t < Tn; ++t) {
      c += dur[b * Tn + t];
      cum[b * Tn + t] = c;
    }
  }
}

__global__ __launch_bounds__(128) void scatter_kernel(
    const float* __restrict__ te, const int* __restrict__ dur,
    const int* __restrict__ cum, float* __restrict__ out, int max_len) {
  const int bt = blockIdx.x;
  const int b = bt >> 10;
  const int end = cum[bt];
  const int d = dur[bt];
  if (d == 0) return;
  const int start = end - d;
  const float4 val = ((const float4*)(te + (size_t)bt * Dn))[threadIdx.x];
  for (int r = start; r < end; ++r) {
    ((float4*)(out + ((size_t)b * max_len + r) * Dn))[threadIdx.x] = val;
  }
}

// ---------------------------------------------------------------------------
extern "C" void kernel_launch(void* const* d_in, const int* in_sizes, int n_in,
                              void* d_out, int out_size, void* d_ws,
                              size_t ws_size, hipStream_t stream) {
  const float* text = (const float*)d_in[0];
  const int* durs = (const int*)d_in[1];
  const float* w1 = (const float*)d_in[2];
  const float* b1 = (const float*)d_in[3];
  const float* g1 = (const float*)d_in[4];
  const float* be1 = (const float*)d_in[5];
  const float* m1 = (const float*)d_in[6];
  const float* v1 = (const float*)d_in[7];
  const float* w2 = (const float*)d_in[8];
  const float* b2 = (const float*)d_in[9];
  const float* g2 = (const float*)d_in[10];
  const float* be2 = (const float*)d_in[11];
  const float* m2 = (const float*)d_in[12];
  const float* v2 = (const float*)d_in[13];
  const float* pw = (const float*)d_in[14];
  const float* pb = (const float*)d_in[15];

  float* ws = (float*)d_ws;
  float* act1 = ws;                                    // [B*T*256]
  float* act2 = act1 + (size_t)Bn * Tn * Hn;           // [B*T*256]
  int* cum = (int*)(act2 + (size_t)Bn * Tn * Hn);      // [B*T]
  float* partials = (float*)(cum + Bn * Tn);           // [128]
  float* wt1 = partials + 128;                         // [256*512*3]
  float* wt2 = wt1 + (size_t)Hn * Dn * 3;              // [256*256*3]

  float* out = (float*)d_out;
  // out = [B*max_len*D expanded | B*T durations | 1 loss]
  const int max_len = (out_size - Bn * Tn - 1) / (Bn * Dn);
  const size_t expN = (size_t)Bn * max_len * Dn;
  float* out_dur = out + expN;
  float* out_loss = out_dur + (size_t)Bn * Tn;

  repack_w_kernel<Dn><<<(Hn * Dn * 3) / 256, 256, 0, stream>>>(w1, wt1);
  repack_w_kernel<Hn><<<(Hn * Hn * 3) / 256, 256, 0, stream>>>(w2, wt2);

  const int blocks = (Bn * Tn) / 32;  // 1024
  conv_bn_relu_wmma<Dn><<<blocks, 256, 0, stream>>>(text, wt1, b1, g1, be1, m1,
                                                    v1, act1);
  conv_bn_relu_wmma<Hn><<<blocks, 256, 0, stream>>>(act1, wt2, b2, g2, be2, m2,
                                                    v2, act2);
  proj_loss_kernel<<<(Bn * Tn) / (8 * 32), 256, 0, stream>>>(act2, pw, pb, durs,
                                                             out_dur, partials);
  loss_final_kernel<<<1, 32, 0, stream>>>(partials, (Bn * Tn) / 256, out_loss);
  zero_kernel<<<1024, 256, 0, stream>>>(out, expN);
  cumsum_kernel<<<1, 32, 0, stream>>>(durs, cum);
  scatter_kernel<<<Bn * Tn, 128, 0, stream>>>(text, durs, cum, out, max_len);
}